// VectorQuantizer_62577673503203
// MI455X (gfx1250) — compile-verified
//
#include <hip/hip_runtime.h>
#include <hip/hip_bf16.h>

// VQ-VAE vector quantizer for MI455X (gfx1250), wave32 + WMMA f32 16x16x4.
//
// inputs:    [32, 128, 8192] f32   (B, D, T)  -> tokens n = b*T + t, feature d
// embedding: [128, 128] f32        (K, D)
// out flat:  [0] loss | [1 .. 33554432] quantized_st (B,D,T) | [33554433] perplexity
//            | [33554434 ..] encodings (N, K) one-hot f32

typedef float v2f __attribute__((ext_vector_type(2)));
typedef float v8f __attribute__((ext_vector_type(8)));

#define DIM      128
#define TLEN     8192
#define KCODES   128
#define NTOK     (32 * 8192)          // 262144 tokens
#define ESTRIDE  132                   // padded LDS row stride (dwords), bank-conflict free
#define Q_OFF    1                     // quantized output offset (floats)
#define PERP_OFF 33554433
#define ENC_OFF  33554434
#define NELEM_F  33554432.0f           // N * D

// dynamic LDS layout (dwords):
//   sEmb : 128*132 = 16896
//   sEn  : 128                 (code norms)
//   sXn  : 64                  (4 waves * 16 row norms)
//   sIdx : 64                  (4 waves * 16 argmin indices)
//   sHist: 128                 (per-WG code histogram)
#define SMEM_DWORDS (16896 + 128 + 64 + 64 + 128)

__global__ void __launch_bounds__(256)
vq_zero(unsigned int* cnt, float* lossAcc) {
    int tid = threadIdx.x;
    if (tid < KCODES) cnt[tid] = 0u;
    if (tid == KCODES) *lossAcc = 0.0f;
}

__global__ void __launch_bounds__(128)
vq_main(const float* __restrict__ in, const float* __restrict__ emb,
        float* __restrict__ out, unsigned int* __restrict__ cnt,
        float* __restrict__ lossAcc)
{
    extern __shared__ float smem[];
    float*        sEmb  = smem;                          // [128][132]
    float*        sEn   = smem + 16896;                  // [128]
    float*        sXn   = smem + 16896 + 128;            // [4][16]
    int*          sIdx  = (int*)(smem + 16896 + 128 + 64);          // [4][16]
    unsigned int* sHist = (unsigned int*)(smem + 16896 + 128 + 128); // [128]

    const int tid = threadIdx.x;
    const int w   = tid >> 5;      // wave id 0..3
    const int l   = tid & 31;      // lane
    const int n   = l & 15;        // row (A) / column (B,C) within 16
    const int kh  = l >> 4;        // K half (0: k%4 in {0,1}, 1: {2,3})

    // ---- stage codebook into LDS (padded rows), zero histogram ----
    for (int i = tid; i < KCODES * DIM; i += 128)
        sEmb[(i >> 7) * ESTRIDE + (i & 127)] = emb[i];
    if (tid < KCODES) sHist[tid] = 0u;
    __syncthreads();

    // ---- per-code squared norms ----
    {
        float s = 0.0f;
        const float* row = sEmb + tid * ESTRIDE;
        #pragma unroll 8
        for (int d = 0; d < DIM; ++d) { float e = row[d]; s += e * e; }
        sEn[tid] = s;
    }
    __syncthreads();

    // ---- this wave's 16-token tile ----
    const int n0 = (blockIdx.x * 4 + w) * 16;   // first token of tile
    const int b  = n0 >> 13;                    // n0 / 8192
    const int t0 = n0 & (TLEN - 1);
    const float* xbase = in + (size_t)b * DIM * TLEN + t0 + n;   // row r = n

    // ---- load A fragments into registers: a[s] = x[row=n][4s + 2*kh + {0,1}] ----
    v2f a[32];
    #pragma unroll
    for (int s = 0; s < 32; ++s) {
        const int k0 = 4 * s + 2 * kh;
        a[s].x = xbase[(size_t)k0 * TLEN];
        a[s].y = xbase[(size_t)(k0 + 1) * TLEN];
    }

    // ---- per-row squared norm (combine the two K-halves) ----
    float pn = 0.0f;
    #pragma unroll
    for (int s = 0; s < 32; ++s) pn += a[s].x * a[s].x + a[s].y * a[s].y;
    pn += __shfl_xor(pn, 16, 32);
    if (l < 16) sXn[w * 16 + n] = pn;           // same-wave LDS ops are in order
    float xn[8];
    #pragma unroll
    for (int v = 0; v < 8; ++v) xn[v] = sXn[w * 16 + kh * 8 + v];  // row M = v + 8*kh

    // ---- distances via WMMA over 8 code tiles, running argmin ----
    float best[8]; int bidx[8];
    #pragma unroll
    for (int v = 0; v < 8; ++v) { best[v] = 3.4e38f; bidx[v] = 0; }

    #pragma unroll
    for (int j = 0; j < 8; ++j) {
        v8f c = {0.f, 0.f, 0.f, 0.f, 0.f, 0.f, 0.f, 0.f};
        const float* brow = sEmb + (16 * j + n) * ESTRIDE + 2 * kh; // B[k][n] = E[16j+n][k]
        #pragma unroll
        for (int s = 0; s < 32; ++s) {
            v2f bf = *(const v2f*)(brow + 4 * s);   // ds_load_b64
            c = __builtin_amdgcn_wmma_f32_16x16x4_f32(
                    false, a[s], false, bf, (short)0, c, false, false);
        }
        const float en = sEn[16 * j + n];
        #pragma unroll
        for (int v = 0; v < 8; ++v) {
            float dist = xn[v] + en - 2.0f * c[v];
            if (dist < best[v]) { best[v] = dist; bidx[v] = 16 * j + n; }
        }
    }

    // ---- cross-lane argmin within each 16-lane half (C cols live per-lane) ----
    #pragma unroll
    for (int v = 0; v < 8; ++v) {
        float bd = best[v]; int bi = bidx[v];
        #pragma unroll
        for (int off = 1; off < 16; off <<= 1) {
            float od = __shfl_xor(bd, off, 32);
            int   oi = __shfl_xor(bi, off, 32);
            if (od < bd || (od == bd && oi < bi)) { bd = od; bi = oi; }
        }
        best[v] = bd; bidx[v] = bi;
    }
    if (n == 0) {   // lanes 0 and 16 hold rows 0-7 / 8-15
        #pragma unroll
        for (int v = 0; v < 8; ++v) sIdx[w * 16 + kh * 8 + v] = bidx[v];
    }

    // ---- quantized store (transposed layout) + fused loss partial ----
    const int myIdx = sIdx[w * 16 + n];          // row r = n of this lane-half
    const float* qrow = sEmb + myIdx * ESTRIDE + 2 * kh;
    float* outQ = out + Q_OFF + (size_t)b * DIM * TLEN + t0 + n;
    float lsum = 0.0f;
    #pragma unroll
    for (int s = 0; s < 32; ++s) {
        v2f e2 = *(const v2f*)(qrow + 4 * s);
        const int k0 = 4 * s + 2 * kh;
        outQ[(size_t)k0 * TLEN]       = e2.x;
        outQ[(size_t)(k0 + 1) * TLEN] = e2.y;
        float d0 = e2.x - a[s].x, d1 = e2.y - a[s].y;
        lsum += d0 * d0 + d1 * d1;
    }
    #pragma unroll
    for (int off = 1; off < 32; off <<= 1) lsum += __shfl_xor(lsum, off, 32);
    if (l == 0) atomicAdd(lossAcc, lsum);
    if (l < 16) atomicAdd(&sHist[(unsigned)sIdx[w * 16 + l]], 1u);

    // ---- one-hot encodings: 16 rows * 128 cols = 1024 float2 per tile ----
    float* enc = out + ENC_OFF + (size_t)n0 * KCODES;
    #pragma unroll
    for (int i = 0; i < 32; ++i) {
        const int f2  = i * 32 + l;          // float2 index in tile, 0..1023
        const int r2  = f2 >> 6;             // 64 float2 per row
        const int c2  = (f2 & 63) * 2;
        const int idx = sIdx[w * 16 + r2];
        v2f val;
        val.x = (c2     == idx) ? 1.0f : 0.0f;
        val.y = (c2 + 1 == idx) ? 1.0f : 0.0f;
        *(v2f*)(enc + (size_t)f2 * 2) = val;
    }

    // ---- flush per-WG histogram ----
    __syncthreads();
    if (tid < KCODES) {
        unsigned int h = sHist[tid];
        if (h) atomicAdd(&cnt[tid], h);
    }
}

__global__ void __launch_bounds__(128)
vq_finalize(const unsigned int* __restrict__ cnt, const float* __restrict__ lossAcc,
            float* __restrict__ out)
{
    __shared__ float red[128];
    const int tid = threadIdx.x;
    float p = (float)cnt[tid] * (1.0f / (float)NTOK);
    red[tid] = p * logf(p + 1e-10f);
    __syncthreads();
    for (int s = 64; s > 0; s >>= 1) {
        if (tid < s) red[tid] += red[tid + s];
        __syncthreads();
    }
    if (tid == 0) {
        out[PERP_OFF] = expf(-red[0]);
        // q_latent_loss + COMMITMENT_COST * e_latent_loss, cost = 1 -> 2 * mean((q-x)^2)
        out[0] = 2.0f * (*lossAcc) / NELEM_F;
    }
}

extern "C" void kernel_launch(void* const* d_in, const int* in_sizes, int n_in,
                              void* d_out, int out_size, void* d_ws, size_t ws_size,
                              hipStream_t stream) {
    const float* in  = (const float*)d_in[0];   // [32,128,8192] f32
    const float* emb = (const float*)d_in[1];   // [128,128] f32
    float* out = (float*)d_out;
    unsigned int* cnt = (unsigned int*)d_ws;                 // 128 uints
    float* lossAcc = (float*)((char*)d_ws + 512);            // 1 float

    vq_zero<<<1, 256, 0, stream>>>(cnt, lossAcc);
    const size_t smemBytes = (size_t)SMEM_DWORDS * 4;        // ~69 KB dynamic LDS
    vq_main<<<NTOK / 64, 128, smemBytes, stream>>>(in, emb, out, cnt, lossAcc);
    vq_finalize<<<1, 128, 0, stream>>>(cnt, lossAcc, out);
}